// OverlapRemovalLayer_10806137717518
// MI455X (gfx1250) — compile-verified
//
#include <hip/hip_runtime.h>
#include <math.h>

#define TWO_PI_F   6.28318530717958647692f
#define INV_2PI_F  0.15915494309189533577f

// ---- gfx1250 async global<->LDS path (guarded; falls back to float4 staging) ----
#if defined(__HIP_DEVICE_COMPILE__) && \
    __has_builtin(__builtin_amdgcn_global_load_async_to_lds_b128) && \
    __has_builtin(__builtin_amdgcn_global_store_async_from_lds_b128)
#define USE_ASYNC_LDS 1
#else
#define USE_ASYNC_LDS 0
#endif

#if USE_ASYNC_LDS
// Builtin params: pointer-to-int4-vector in AS1 (global) / AS3 (LDS), non-const.
typedef int v4i_async __attribute__((vector_size(16)));
typedef __attribute__((address_space(1))) v4i_async* g4ptr_t;
typedef __attribute__((address_space(3))) v4i_async* l4ptr_t;
#endif

__device__ __forceinline__ void wait_asynccnt0() {
#if USE_ASYNC_LDS
#if __has_builtin(__builtin_amdgcn_s_wait_asynccnt)
    __builtin_amdgcn_s_wait_asynccnt(0);
#else
    asm volatile("s_wait_asynccnt 0" ::: "memory");
#endif
#endif
}

// Remove rule: zero target j if ANY remover i has masked dR < thr.
// Reference semantics: inactive objects (comp0 <= 0) use sentinel eta=phi=1e6 on
// BOTH sides; two inactive objects give dR=0 < thr (inactive remover zeroes
// inactive target), mixed active/inactive gives dR ~ 1e6 (no kill).
// dphi wrap: dp0 - 2pi*rndne(dp0/2pi) == (mod(dp0+pi,2pi)-pi) up to sign at the
// exact +/-pi boundary, which is irrelevant after squaring. For the one-sentinel
// case (dp0 ~ +/-1e6) the reduction is inexact by ~0.1, but de^2 ~ 1e12 dominates.
template <int RB, int NR, int TB, int NT>
__device__ __forceinline__ void rule(float* o, float thr) {
    const float t2 = thr * thr;
    float re[NR], rp[NR], te[NT], tp[NT];
#pragma unroll
    for (int i = 0; i < NR; ++i) {
        const bool a = o[(RB + i) * 3 + 0] > 0.f;
        re[i] = a ? o[(RB + i) * 3 + 1] : 1e6f;
        rp[i] = a ? o[(RB + i) * 3 + 2] : 1e6f;
    }
#pragma unroll
    for (int j = 0; j < NT; ++j) {
        const bool a = o[(TB + j) * 3 + 0] > 0.f;
        te[j] = a ? o[(TB + j) * 3 + 1] : 1e6f;
        tp[j] = a ? o[(TB + j) * 3 + 2] : 1e6f;
    }
    bool kill[NT];
#pragma unroll
    for (int j = 0; j < NT; ++j) {
        float smin = 1e30f;
#pragma unroll
        for (int i = 0; i < NR; ++i) {
            const float de  = re[i] - te[j];
            const float dp0 = rp[i] - tp[j];
            const float dp  = fmaf(-TWO_PI_F, rintf(dp0 * INV_2PI_F), dp0);
            const float s   = fmaf(de, de, dp * dp);
            smin = fminf(smin, s);
        }
        kill[j] = smin < t2;
    }
#pragma unroll
    for (int j = 0; j < NT; ++j) {
        if (kill[j]) {
            o[(TB + j) * 3 + 0] = 0.f;
            o[(TB + j) * 3 + 1] = 0.f;
            o[(TB + j) * 3 + 2] = 0.f;
        }
    }
}

__global__ __launch_bounds__(256)
void overlap_removal_kernel(const float4* __restrict__ gin,
                            float4* __restrict__ gout, int N) {
    constexpr int EPB    = 256;  // events per block (1 per thread)
    constexpr int QPE    = 12;   // float4 quads per event (48 floats)
    constexpr int STRIDE = 52;   // padded floats per event in LDS (16B-aligned, low bank conflicts)
    __shared__ __align__(16) float smem[EPB * STRIDE];

    const int tid      = threadIdx.x;
    const int blockEv0 = blockIdx.x * EPB;
    const long long qbase = (long long)blockIdx.x * (EPB * QPE);

    // ---------------- Stage IN: fully coalesced 128-bit global -> LDS (async DMA) ----------------
#pragma unroll
    for (int k = 0; k < QPE; ++k) {
        const int g  = tid + k * EPB;          // consecutive tid -> consecutive global float4
        const int ev = g / QPE;
        const int q  = g - ev * QPE;
        if (blockEv0 + ev < N) {
#if USE_ASYNC_LDS
            __builtin_amdgcn_global_load_async_to_lds_b128(
                (g4ptr_t)(void*)(gin + qbase + g),
                (l4ptr_t)(void*)(&smem[ev * STRIDE + q * 4]),
                0, 0);
#else
            *reinterpret_cast<float4*>(&smem[ev * STRIDE + q * 4]) = gin[qbase + g];
#endif
        }
    }
    wait_asynccnt0();
    __syncthreads();

    // ---------------- Compute: one event per thread, all in registers ----------------
    if (blockEv0 + tid < N) {
        float o[48];
        float4* ov = reinterpret_cast<float4*>(o);
#pragma unroll
        for (int k = 0; k < QPE; ++k)
            ov[k] = *reinterpret_cast<const float4*>(&smem[tid * STRIDE + k * 4]);

        // Layout: jets 0..5, electrons 6..8, muons 9..11, photons 12..14, met 15
        rule<6, 3, 9, 3>(o, 0.2f);   // muons    = rm(electrons, muons,    0.2)
        rule<6, 3, 12, 3>(o, 0.4f);  // photons  = rm(electrons, photons,  0.4)
        rule<9, 3, 12, 3>(o, 0.4f);  // photons  = rm(muons,     photons,  0.4)
        rule<6, 3, 0, 6>(o, 0.2f);   // jets     = rm(electrons, jets,     0.2)
        rule<0, 6, 6, 3>(o, 0.4f);   // electrons= rm(jets,      electrons,0.4)
        rule<9, 3, 0, 6>(o, 0.2f);   // jets     = rm(muons,     jets,     0.2)
        rule<0, 6, 9, 3>(o, 0.4f);   // muons    = rm(jets,      muons,    0.4)
        rule<0, 6, 12, 3>(o, 0.4f);  // photons  = rm(jets,      photons,  0.4)
        // met (object 15) passes through untouched.

#pragma unroll
        for (int k = 0; k < QPE; ++k)
            *reinterpret_cast<float4*>(&smem[tid * STRIDE + k * 4]) = ov[k];
    }
    __syncthreads();

    // ---------------- Stage OUT: fully coalesced 128-bit LDS -> global (async DMA) ----------------
#pragma unroll
    for (int k = 0; k < QPE; ++k) {
        const int g  = tid + k * EPB;
        const int ev = g / QPE;
        const int q  = g - ev * QPE;
        if (blockEv0 + ev < N) {
#if USE_ASYNC_LDS
            __builtin_amdgcn_global_store_async_from_lds_b128(
                (g4ptr_t)(void*)(gout + qbase + g),
                (l4ptr_t)(void*)(&smem[ev * STRIDE + q * 4]),
                0, 0);
#else
            gout[qbase + g] = *reinterpret_cast<const float4*>(&smem[ev * STRIDE + q * 4]);
#endif
        }
    }
    // S_ENDPGM performs an implicit wait-idle, covering outstanding ASYNCcnt stores.
}

extern "C" void kernel_launch(void* const* d_in, const int* in_sizes, int n_in,
                              void* d_out, int out_size, void* d_ws, size_t ws_size,
                              hipStream_t stream) {
    (void)n_in; (void)d_ws; (void)ws_size; (void)out_size;
    const float4* x = reinterpret_cast<const float4*>(d_in[0]);
    float4* out     = reinterpret_cast<float4*>(d_out);
    const int N     = in_sizes[0] / 48;           // events (16 objects x 3 floats)
    const int blocks = (N + 255) / 256;
    overlap_removal_kernel<<<dim3(blocks), dim3(256), 0, stream>>>(x, out, N);
}